// SparseConv2d_12867722018963
// MI455X (gfx1250) — compile-verified
//
#include <hip/hip_runtime.h>

// fp32 WMMA fragment types for V_WMMA_F32_16X16X4_F32 (wave32):
//   A: 16x4 f32 -> 2 VGPRs/lane, B: 4x16 f32 -> 2 VGPRs/lane, C/D: 16x16 f32 -> 8 VGPRs/lane
typedef __attribute__((ext_vector_type(2))) float v2f;
typedef __attribute__((ext_vector_type(8))) float v8f;

#define N_BATCH   32
#define C_IN      64
#define C_OUT     64
#define H_DIM     128
#define W_DIM     128
#define KSUM      (C_IN * 9)      // 576 reduction length
#define LW_STRIDE 578             // weight LDS row stride (floats): even -> 8B-aligned b64 loads

// Input halo tile for a 64-wide w strip: 3 rows x 66 cols, channel-PAIR interleaved:
//   li[pair][pos][2], pos = row*66+col, 198 positions -> 396 floats; pad to 416 (416%64==32)
#define LI_COLS   66
#define LI_POS    (3 * LI_COLS)          // 198
#define LI_PSTRIDE 416
#define LW_FLOATS (32 * LW_STRIDE)
#define LI_FLOATS ((C_IN / 2) * LI_PSTRIDE)
#define SMEM_BYTES ((size_t)(LW_FLOATS + LI_FLOATS) * sizeof(float))  // ~124 KB -> 2 blocks/WGP

__global__ __launch_bounds__(128) void sparse_conv2d_wmma(
    const float* __restrict__ x,       // (32, 64, 128, 128)
    const float* __restrict__ weight,  // (64, 64, 3, 3)
    const float* __restrict__ bias,    // (64,)
    float* __restrict__ out)           // (32, 64, 128, 128)
{
  extern __shared__ float smem[];
  float* lw = smem;              // lw[m][k], m in 0..31 (och in group), k = (dh*3+dw)*64 + c
  float* li = smem + LW_FLOATS;  // li[c>>1][row*66+col][c&1]

  const int tid     = threadIdx.x;
  const int wBase   = (blockIdx.x >> 1) * 64;   // which 64-wide w strip
  const int ochBase = (blockIdx.x & 1) * 32;    // which 32-channel output group
  const int h       = blockIdx.y;
  const int n       = blockIdx.z;

  // ---- Stage weights into LDS, transposed to k = (dh*3+dw)*64 + c ordering ----
  const float* wsrc = weight + (size_t)ochBase * KSUM;
  for (int idx = tid; idx < 32 * KSUM; idx += 128) {
    int m    = idx / KSUM;
    int r    = idx - m * KSUM;
    int c    = r / 9;
    int dhdw = r - c * 9;
    lw[m * LW_STRIDE + dhdw * 64 + c] = wsrc[idx];
  }

  // ---- Stage input halo tile (64ch x 3 rows x 66 cols), channel-pair interleaved ----
  for (int idx = tid; idx < C_IN * LI_POS; idx += 128) {
    int c   = idx / LI_POS;
    int r   = idx - c * LI_POS;       // pos = row*66 + col
    int row = r / LI_COLS;
    int col = r - row * LI_COLS;
    int gh  = h - 1 + row;
    int gw  = wBase - 1 + col;
    float v = 0.0f;
    if ((unsigned)gh < (unsigned)H_DIM && (unsigned)gw < (unsigned)W_DIM)
      v = x[(((size_t)n * C_IN + c) * H_DIM + gh) * W_DIM + gw];
    li[(c >> 1) * LI_PSTRIDE + r * 2 + (c & 1)] = v;
  }

  __syncthreads();

  // ---- Per-wave: 2 M-tiles (32 och) x 1 N-tile (16 w); B fragment reused by 2 WMMAs ----
  const int lane  = tid & 31;
  const int wave  = tid >> 5;   // nTile: which 16-wide w sub-strip (0..3)
  const int mrow  = lane & 15;  // M row (A) / N col (B,D) index
  const int ksel  = lane >> 4;  // half-wave: K pair {k0,k0+1} vs {k0+2,k0+3}

  // A fragments: rows mrow (tile 0) and 16+mrow (tile 1), k0 + 2*ksel
  const float* A0 = lw + mrow * LW_STRIDE + 2 * ksel;
  const float* A1 = A0 + 16 * LW_STRIDE;
  // B fragment: channels (c0+2*ksel, c0+2*ksel+1) adjacent dwords at
  //   li[(c0>>1)+ksel][dh*66 + dw + wave*16 + mrow][0..1]
  const float* Bbase = li + (size_t)ksel * LI_PSTRIDE + (wave * 16 + mrow) * 2;

  v8f acc0 = {}, acc1 = {};
  #pragma unroll
  for (int dh = 0; dh < 3; ++dh) {
    #pragma unroll
    for (int dw = 0; dw < 3; ++dw) {
      const int koff = (dh * 3 + dw) * 64;
      const float* Bq = Bbase + (dh * LI_COLS + dw) * 2;
      #pragma unroll
      for (int c0 = 0; c0 < 64; c0 += 4) {
        v2f b  = *(const v2f*)(Bq + (c0 >> 1) * LI_PSTRIDE);  // ds_load_b64, shared by 2 WMMAs
        v2f a0 = *(const v2f*)(A0 + koff + c0);               // ds_load_b64
        v2f a1 = *(const v2f*)(A1 + koff + c0);               // ds_load_b64
        acc0 = __builtin_amdgcn_wmma_f32_16x16x4_f32(
            false, a0, false, b, (short)0, acc0, false, false);
        acc1 = __builtin_amdgcn_wmma_f32_16x16x4_f32(
            false, a1, false, b, (short)0, acc1, false, false);
      }
    }
  }

  // ---- Epilogue: D layout is VGPR r -> M = r + 8*ksel, lane -> N = mrow ----
  const int wOut = wBase + wave * 16 + mrow;
  const size_t outSpatial = ((size_t)n * C_OUT) * H_DIM * W_DIM + (size_t)h * W_DIM + wOut;
  const int o0 = ochBase + ksel * 8;        // tile 0 channel base
  #pragma unroll
  for (int r = 0; r < 8; ++r) {
    const int oA = o0 + r;
    const int oB = o0 + 16 + r;
    out[outSpatial + (size_t)oA * H_DIM * W_DIM] = acc0[r] + bias[oA];
    out[outSpatial + (size_t)oB * H_DIM * W_DIM] = acc1[r] + bias[oB];
  }
}

extern "C" void kernel_launch(void* const* d_in, const int* in_sizes, int n_in,
                              void* d_out, int out_size, void* d_ws, size_t ws_size,
                              hipStream_t stream) {
  const float* x = (const float*)d_in[0];
  const float* w = (const float*)d_in[1];
  const float* b = (const float*)d_in[2];
  float* y = (float*)d_out;
  (void)in_sizes; (void)n_in; (void)out_size; (void)d_ws; (void)ws_size;

  dim3 grid((W_DIM / 64) * 2, H_DIM, N_BATCH);  // (w-strips x och-groups, H, N)
  dim3 block(128);                               // 4 wave32s, each: 32 och x 16 w
  sparse_conv2d_wmma<<<grid, block, SMEM_BYTES, stream>>>(x, w, b, y);
}